// RobustCombiner_21534966022154
// MI455X (gfx1250) — compile-verified
//
#include <hip/hip_runtime.h>

#define VOCAB 42024
#define KNN   32
#define RK    6
#define FDIM  64
#define TOKB  128   // tokens per block
#define THREADS 128

typedef float v2f __attribute__((ext_vector_type(2)));
typedef float v4f __attribute__((ext_vector_type(4)));
typedef float v8f __attribute__((ext_vector_type(8)));

// ---------------------------------------------------------------------------
// Kernel 1: nontemporal zero fill of the 688MB output (bandwidth bound part).
// ---------------------------------------------------------------------------
__global__ void __launch_bounds__(256) rc_zero_kernel(float* __restrict__ out, long n) {
    long n4 = n >> 2;
    long i = (long)blockIdx.x * blockDim.x + threadIdx.x;
    long stride = (long)gridDim.x * blockDim.x;
    v4f z = {0.f, 0.f, 0.f, 0.f};
    v4f* __restrict__ o4 = (v4f*)out;
    for (long k = i; k < n4; k += stride)
        __builtin_nontemporal_store(z, &o4[k]);
    // tail (out_size is divisible by 4 here, but stay safe)
    for (long k = (n4 << 2) + i; k < n; k += stride)
        out[k] = 0.f;
}

// ---------------------------------------------------------------------------
// Kernel 2: per-token features -> WMMA logits -> softmax combine -> scatter.
// ---------------------------------------------------------------------------
__global__ void __launch_bounds__(THREADS) rc_combiner_kernel(
    const int*   __restrict__ vals,       // [4096,32]
    const float* __restrict__ distances,  // [4096,32]
    const float* __restrict__ Wk,         // [64,6] row-major
    const float* __restrict__ bk,         // [6]
    const float* __restrict__ Wt,         // [64,1]
    const float* __restrict__ bt,         // [1]
    float*       __restrict__ out)        // [4096, VOCAB]
{
    __shared__ int   vals_s [TOKB * KNN];   // 16 KB
    __shared__ float feats_s[TOKB * FDIM];  // 32 KB : [dist(32) | counts(32)]
    __shared__ float logit_s[TOKB * 8];     //  4 KB : 7 logits (padded)

    const int t    = threadIdx.x;
    const int gtok = blockIdx.x * TOKB + t;

    // ---- Phase 1: load vals/dist, distinct-nonzero prefix counts ----
    {
        const int*   vp = vals      + (size_t)gtok * KNN;
        const float* dp = distances + (size_t)gtok * KNN;
        #pragma unroll
        for (int j = 0; j < KNN; ++j) {
            vals_s [t * KNN  + j] = vp[j];
            feats_s[t * FDIM + j] = dp[j];
        }
        // same-wave LDS ops are in-order: own reads below see own writes
        int c = 0;
        for (int j = 0; j < KNN; ++j) {
            int vj = vals_s[t * KNN + j];
            int isnew = (vj != 0) ? 1 : 0;
            for (int i = 0; i < j; ++i)
                if (vals_s[t * KNN + i] == vj) isnew = 0;
            c += isnew;
            feats_s[t * FDIM + KNN + j] = (float)c;
        }
    }
    __syncthreads();

    // ---- Phase 2: logits = feats @ [Wk|Wt] via V_WMMA_F32_16X16X4_F32 ----
    {
        const int lane = t & 31;
        const int wave = t >> 5;
        const int row  = lane & 15;
        const int hi   = lane >> 4;   // half-wave selects K parity (0,1 vs 2,3)

        // B fragments: Wfull[k][n], n = row; col 6 = Wt, cols 7..15 = 0
        float b0[16], b1[16];
        #pragma unroll
        for (int s = 0; s < 16; ++s) {
            int k0 = 4 * s + 2 * hi;
            float x0 = 0.f, x1 = 0.f;
            if (row < RK)       { x0 = Wk[k0 * RK + row]; x1 = Wk[(k0 + 1) * RK + row]; }
            else if (row == RK) { x0 = Wt[k0];            x1 = Wt[k0 + 1]; }
            b0[s] = x0; b1[s] = x1;
        }

        #pragma unroll
        for (int half = 0; half < 2; ++half) {
            const int tokBase = wave * 32 + half * 16;
            v8f acc = {};
            #pragma unroll
            for (int s = 0; s < 16; ++s) {
                int k0 = 4 * s + 2 * hi;
                v2f a, b;
                a.x = feats_s[(tokBase + row) * FDIM + k0];
                a.y = feats_s[(tokBase + row) * FDIM + k0 + 1];
                b.x = b0[s];
                b.y = b1[s];
                acc = __builtin_amdgcn_wmma_f32_16x16x4_f32(
                    false, a, false, b, (short)0, acc, false, false);
            }
            // C layout: vgpr i -> M = i (+8 for hi half), N = lane%16
            #pragma unroll
            for (int i = 0; i < 8; ++i) {
                int m = hi ? (i + 8) : i;
                if (row < 7) {
                    float bias = (row < RK) ? bk[row] : bt[0];
                    logit_s[(tokBase + m) * 8 + row] = acc[i] + bias;
                }
            }
        }
    }
    __syncthreads();

    // ---- Phase 3: softmax/sigmoid, masked knn softmaxes, scatter ----
    {
        float l[7];
        #pragma unroll
        for (int n = 0; n < 7; ++n) l[n] = logit_s[t * 8 + n];

        float mx = l[0];
        #pragma unroll
        for (int n = 1; n < RK; ++n) mx = fmaxf(mx, l[n]);
        float kp[RK]; float se = 0.f;
        #pragma unroll
        for (int n = 0; n < RK; ++n) { kp[n] = __expf(l[n] - mx); se += kp[n]; }
        float inv = 1.f / se;
        #pragma unroll
        for (int n = 0; n < RK; ++n) kp[n] *= inv;

        float temp = 1.f / (1.f + __expf(-l[6]));   // sigmoid
        float invT = 1.f / temp;

        float w[KNN];
        #pragma unroll
        for (int c = 0; c < KNN; ++c) w[c] = 0.f;

        const float* df = &feats_s[t * FDIM];
        #pragma unroll
        for (int r = 0; r < RK; ++r) {
            const int pw = (1 << r) - 1;   // pows = 2^r - 1
            float dmx = -3.4e38f;
            #pragma unroll
            for (int c = 0; c < KNN; ++c) {
                float dd = -df[c] * ((c <= pw) ? 1.0f : 1000.0f) * invT;
                dmx = fmaxf(dmx, dd);
            }
            float ssum = 0.f;
            #pragma unroll
            for (int c = 0; c < KNN; ++c) {
                float dd = -df[c] * ((c <= pw) ? 1.0f : 1000.0f) * invT;
                ssum += __expf(dd - dmx);
            }
            float scale = kp[r] / ssum;
            #pragma unroll
            for (int c = 0; c < KNN; ++c) {
                float dd = -df[c] * ((c <= pw) ? 1.0f : 1000.0f) * invT;
                w[c] += scale * __expf(dd - dmx);
            }
        }

        float* orow = out + (size_t)gtok * VOCAB;
        #pragma unroll
        for (int c = 0; c < KNN; ++c)
            atomicAdd(orow + vals_s[t * KNN + c], w[c]);
    }
}

// ---------------------------------------------------------------------------
extern "C" void kernel_launch(void* const* d_in, const int* in_sizes, int n_in,
                              void* d_out, int out_size, void* d_ws, size_t ws_size,
                              hipStream_t stream) {
    const int*   vals      = (const int*)  d_in[0];
    const float* distances = (const float*)d_in[1];
    const float* Wk        = (const float*)d_in[2];
    const float* bk        = (const float*)d_in[3];
    const float* Wt        = (const float*)d_in[4];
    const float* bt        = (const float*)d_in[5];
    float* out = (float*)d_out;

    // Phase A: zero the 688MB output at full HBM bandwidth (NT stores).
    rc_zero_kernel<<<8192, 256, 0, stream>>>(out, (long)out_size);

    // Phase B: compute + scatter (4096 tokens, 128 per block).
    int n_tok  = in_sizes[0] / KNN;      // 4096
    int blocks = (n_tok + TOKB - 1) / TOKB;
    rc_combiner_kernel<<<blocks, THREADS, 0, stream>>>(
        vals, distances, Wk, bk, Wt, bt, out);
}